// SSMBranch_18949395710586
// MI455X (gfx1250) — compile-verified
//
#include <hip/hip_runtime.h>
#include <hip/hip_bf16.h>

// ---------------------------------------------------------------------------
// SSM branch (Mamba-like) for MI455X / gfx1250, fp32 end-to-end.
// GEMMs: V_WMMA_F32_16X16X4_F32, 4 accumulators/wave, weight slab in LDS.
// Scan: chunked (associative) 3-pass formulation for parallelism.
// ---------------------------------------------------------------------------

#define M_TOK   16384   // B*N
#define SEQ_N   4096
#define BATCH   4
#define D_MODEL 512
#define D_INNER 1024
#define D_STATE 16
#define DT_RANK 16
#define N_OUT   256
#define CHUNK   256
#define NCHUNK  16      // SEQ_N / CHUNK

typedef __attribute__((ext_vector_type(2))) float v2f;
typedef __attribute__((ext_vector_type(8))) float v8f;

// ---- WMMA f32 16x16x4: D = A(16x4) x B(4x16) + C(16x16) ----
__device__ __forceinline__ v8f wmma4(v2f a, v2f b, v8f c) {
    return __builtin_amdgcn_wmma_f32_16x16x4_f32(
        /*neg_a=*/false, a, /*neg_b=*/false, b,
        /*c_mod=*/(short)0, c, /*reuse_a=*/false, /*reuse_b=*/false);
}

// A fragment (16x4). ISA: lanes 0-15 -> M=lane, K={0,1}; lanes 16-31 -> K={2,3}.
__device__ __forceinline__ v2f load_a(const float* __restrict__ A, int lda,
                                      int mrow, int k0, int lane) {
    int m = lane & 15, h = lane >> 4;
    const float* p = A + (size_t)(mrow + m) * lda + (k0 + 2 * h);
    v2f a; a.x = p[0]; a.y = p[1];
    return a;
}

// B fragment (4x16) from global. VGPR j holds K=j (lanes 0-15) / K=j+2 (16-31).
__device__ __forceinline__ v2f load_b(const float* __restrict__ B, int ldb,
                                      int k0, int nb, int lane) {
    int n = lane & 15, h = lane >> 4;
    const float* p = B + (size_t)(k0 + 2 * h) * ldb + (nb + n);
    v2f b; b.x = p[0]; b.y = p[ldb];
    return b;
}

// B fragment from LDS pair layout: pair p=k/2 stores [n][k&1] contiguously,
// so each lane's two K values are one ds_load_b64.
__device__ __forceinline__ v2f load_b_lds(const float* Bs2, int k0, int lane) {
    int n = lane & 15, h = lane >> 4;
    const float* p = Bs2 + ((k0 >> 1) + h) * 32 + n * 2;
    v2f b; b.x = p[0]; b.y = p[1];
    return b;
}

__device__ __forceinline__ float silu_f(float v) {
    return v / (1.0f + __expf(-v));
}

// ---------------------------------------------------------------------------
// K1: x(16384x512) @ W_in(512x2048) + b_in, SiLU, split into x_main / gate.
// Block = 8 waves, 512 rows x 16 cols. W slab (512x16 = 32KB) staged in LDS.
// Each wave: 4 M-tiles, 4 accumulators, shared B fragment.
// ---------------------------------------------------------------------------
__global__ void __launch_bounds__(256)
k_gemm_in_silu(const float* __restrict__ x, const float* __restrict__ W,
               const float* __restrict__ bias,
               float* __restrict__ xm, float* __restrict__ g) {
    __shared__ float Bs2[D_MODEL * 16];            // 32 KB, pair layout
    const int tid  = threadIdx.x;
    const int lane = tid & 31, wave = tid >> 5;
    const int nb   = (blockIdx.x & 127) * 16;      // N tile (16 cols of 2048)
    const int mBlk = blockIdx.x >> 7;              // 0..31 (512 rows each)

    for (int i = tid; i < D_MODEL * 16; i += 256) {
        int k = i >> 4, n = i & 15;
        Bs2[(k >> 1) * 32 + n * 2 + (k & 1)] =
            W[(size_t)k * (2 * D_INNER) + nb + n];
    }
    __syncthreads();

    const int mrow = mBlk * 512 + wave * 64;
    const int m = lane & 15, h = lane >> 4, n = lane & 15;
    v8f c0 = {}, c1 = {}, c2 = {}, c3 = {};
    for (int k0 = 0; k0 < D_MODEL; k0 += 4) {
        if (k0 + 64 < D_MODEL)
            __builtin_prefetch(&x[(size_t)(mrow + m) * D_MODEL + k0 + 64], 0, 3);
        v2f b  = load_b_lds(Bs2, k0, lane);
        v2f a0 = load_a(x, D_MODEL, mrow,      k0, lane);
        v2f a1 = load_a(x, D_MODEL, mrow + 16, k0, lane);
        v2f a2 = load_a(x, D_MODEL, mrow + 32, k0, lane);
        v2f a3 = load_a(x, D_MODEL, mrow + 48, k0, lane);
        c0 = wmma4(a0, b, c0);
        c1 = wmma4(a1, b, c1);
        c2 = wmma4(a2, b, c2);
        c3 = wmma4(a3, b, c3);
    }
    const float bv = bias[nb + n];
    float* dst = (nb < D_INNER) ? xm : g;
    const int col = ((nb < D_INNER) ? nb : nb - D_INNER) + n;
    v8f cc[4] = {c0, c1, c2, c3};
#pragma unroll
    for (int ti = 0; ti < 4; ++ti)
#pragma unroll
        for (int r = 0; r < 8; ++r) {
            float v = cc[ti][r] + bv;
            dst[(size_t)(mrow + ti * 16 + r + 8 * h) * D_INNER + col] = silu_f(v);
        }
}

// ---------------------------------------------------------------------------
// K2: skinny projections: dt_low = xm@W_dt1 + b_dt1, Bp = xm@W_B, Cp = xm@W_C.
// 1024 M-tiles x 3 targets; K loop over 1024; weights are L2-resident (64KB).
// ---------------------------------------------------------------------------
__global__ void __launch_bounds__(256)
k_proj16(const float* __restrict__ xm,
         const float* __restrict__ Wdt1, const float* __restrict__ bdt1,
         const float* __restrict__ WB, const float* __restrict__ WC,
         float* __restrict__ dtlow, float* __restrict__ Bp, float* __restrict__ Cp) {
    const int lane = threadIdx.x & 31;
    const int tile = blockIdx.x * 8 + (threadIdx.x >> 5); // 0..3071
    const int tm = tile & 1023;
    const int which = tile >> 10;                          // 0:dt1 1:B 2:C
    const float* Bmat = (which == 0) ? Wdt1 : ((which == 1) ? WB : WC);
    float* dst = (which == 0) ? dtlow : ((which == 1) ? Bp : Cp);
    const int mrow = tm * 16;
    v8f c = {};
    for (int k0 = 0; k0 < D_INNER; k0 += 4) {
        v2f a = load_a(xm, D_INNER, mrow, k0, lane);
        v2f b = load_b(Bmat, D_STATE, k0, 0, lane);
        c = wmma4(a, b, c);
    }
    const int n = lane & 15, h = lane >> 4;
    const float bv = (which == 0) ? bdt1[n] : 0.0f;
#pragma unroll
    for (int r = 0; r < 8; ++r)
        dst[(size_t)(mrow + r + 8 * h) * D_STATE + n] = c[r] + bv;
}

// ---------------------------------------------------------------------------
// K3: dt = clip(softplus(dt_low @ W_dt2 + b_dt2), 1e-6, 10). K=16 (4 WMMAs).
// ---------------------------------------------------------------------------
__global__ void __launch_bounds__(256)
k_dt_proj(const float* __restrict__ dtlow, const float* __restrict__ W2,
          const float* __restrict__ b2, float* __restrict__ dt) {
    const int lane = threadIdx.x & 31;
    const int tile = blockIdx.x * 8 + (threadIdx.x >> 5); // 1024 x 64 tiles
    const int mrow = (tile >> 6) * 16;
    const int nb   = (tile & 63) * 16;
    v8f c = {};
#pragma unroll
    for (int k0 = 0; k0 < DT_RANK; k0 += 4) {
        v2f a = load_a(dtlow, DT_RANK, mrow, k0, lane);
        v2f b = load_b(W2, D_INNER, k0, nb, lane);
        c = wmma4(a, b, c);
    }
    const int n = lane & 15, h = lane >> 4;
    const float bv = b2[nb + n];
#pragma unroll
    for (int r = 0; r < 8; ++r) {
        float v = c[r] + bv;
        float sp = (v > 20.0f) ? v : log1pf(__expf(v));      // softplus
        sp = fminf(fmaxf(sp, 1e-6f), 10.0f);                 // clip
        dt[(size_t)(mrow + r + 8 * h) * D_INNER + nb + n] = sp;
    }
}

// ---------------------------------------------------------------------------
// Chunked scan. Per (dir, b, d) channel the update is affine per state s:
//   h' = A_bar*h + (B_t*dt)*u ,  A_bar = exp(clip(A*dt,-5,5))
// Pass 1: per-chunk affine summary (P = prod A_bar, Q = accumulated input).
// Pass 2: sequentially compose 16 chunk maps -> chunk entry states.
// Pass 3: re-scan chunks from entry state, emit y = C.h + D*u.
// ---------------------------------------------------------------------------
__global__ void __launch_bounds__(256)
k_scan_chunks(const float* __restrict__ xm, const float* __restrict__ dt,
              const float* __restrict__ Bp, const float* __restrict__ A_log,
              float* __restrict__ Pbuf, float* __restrict__ Qbuf) {
    const int tid = blockIdx.x * 256 + threadIdx.x;  // 131072 threads
    const int chunk = tid & 15;
    const int d = (tid >> 4) & 1023;
    const int b = (tid >> 14) & 3;
    const int dir = tid >> 16;
    float A[D_STATE], P[D_STATE], Q[D_STATE];
#pragma unroll
    for (int s = 0; s < D_STATE; ++s) {
        A[s] = -__expf(A_log[d * D_STATE + s]);
        P[s] = 1.0f; Q[s] = 0.0f;
    }
    const int j0 = chunk * CHUNK;
    for (int j = j0; j < j0 + CHUNK; ++j) {
        const int t = dir ? (SEQ_N - 1 - j) : j;
        const size_t row = (size_t)b * SEQ_N + t;
        const size_t idx = row * D_INNER + d;
        const float u = xm[idx], dv = dt[idx];
        const float du = dv * u;
        const float* br = Bp + row * D_STATE;
#pragma unroll
        for (int s = 0; s < D_STATE; ++s) {
            float ar = fminf(fmaxf(A[s] * dv, -5.0f), 5.0f);
            float ab = __expf(ar);
            P[s] *= ab;
            Q[s] = __builtin_fmaf(ab, Q[s], br[s] * du);
        }
    }
    const size_t base = (((size_t)((dir * BATCH + b) * D_INNER + d)) * NCHUNK + chunk) * D_STATE;
#pragma unroll
    for (int s = 0; s < D_STATE; ++s) { Pbuf[base + s] = P[s]; Qbuf[base + s] = Q[s]; }
}

__global__ void __launch_bounds__(256)
k_scan_compose(const float* __restrict__ Pbuf, const float* __restrict__ Qbuf,
               float* __restrict__ Hst) {
    const int tid = blockIdx.x * 256 + threadIdx.x;  // 8192 threads: (dir,b,d)
    float h[D_STATE];
#pragma unroll
    for (int s = 0; s < D_STATE; ++s) h[s] = 0.0f;
    const size_t chbase = (size_t)tid * NCHUNK * D_STATE;
    for (int c = 0; c < NCHUNK; ++c) {
        const size_t o = chbase + (size_t)c * D_STATE;
#pragma unroll
        for (int s = 0; s < D_STATE; ++s) {
            Hst[o + s] = h[s];                               // entry state of chunk c
            h[s] = __builtin_fmaf(Pbuf[o + s], h[s], Qbuf[o + s]);
        }
    }
}

__global__ void __launch_bounds__(256)
k_scan_apply(const float* __restrict__ xm, const float* __restrict__ dt,
             const float* __restrict__ Bp, const float* __restrict__ Cp,
             const float* __restrict__ A_log, const float* __restrict__ Dp,
             const float* __restrict__ Hst,
             float* __restrict__ yf, float* __restrict__ yb) {
    const int tid = blockIdx.x * 256 + threadIdx.x;  // 131072 threads
    const int chunk = tid & 15;
    const int d = (tid >> 4) & 1023;
    const int b = (tid >> 14) & 3;
    const int dir = tid >> 16;
    float A[D_STATE], h[D_STATE];
    const size_t base = (((size_t)((dir * BATCH + b) * D_INNER + d)) * NCHUNK + chunk) * D_STATE;
#pragma unroll
    for (int s = 0; s < D_STATE; ++s) {
        A[s] = -__expf(A_log[d * D_STATE + s]);
        h[s] = Hst[base + s];
    }
    const float Dv = Dp[d];
    float* __restrict__ y = dir ? yb : yf;
    const int j0 = chunk * CHUNK;
    for (int j = j0; j < j0 + CHUNK; ++j) {
        const int t = dir ? (SEQ_N - 1 - j) : j;
        const size_t row = (size_t)b * SEQ_N + t;
        const size_t idx = row * D_INNER + d;
        const float u = xm[idx], dv = dt[idx];
        const float du = dv * u;
        const float* br = Bp + row * D_STATE;
        const float* cr = Cp + row * D_STATE;
        float acc = 0.0f;
#pragma unroll
        for (int s = 0; s < D_STATE; ++s) {
            float ar = fminf(fmaxf(A[s] * dv, -5.0f), 5.0f);
            float ab = __expf(ar);
            h[s] = __builtin_fmaf(ab, h[s], br[s] * du);
            acc = __builtin_fmaf(cr[s], h[s], acc);
        }
        y[idx] = acc + Dv * u;
    }
}

// ---------------------------------------------------------------------------
// K5: out = ((yf+yb) * gate) @ W_out(1024x256). Gating fused into A loads.
// Block = 8 waves, 512 rows x 16 cols; K=1024 staged through LDS in 2 phases
// of 32KB each. Each wave: 4 M-tiles, 4 accumulators.
// ---------------------------------------------------------------------------
__global__ void __launch_bounds__(256)
k_gemm_out(const float* __restrict__ yf, const float* __restrict__ yb,
           const float* __restrict__ g, const float* __restrict__ W,
           float* __restrict__ out) {
    __shared__ float Bs2[512 * 16];                // 32 KB, pair layout
    const int tid  = threadIdx.x;
    const int lane = tid & 31, wave = tid >> 5;
    const int nb   = (blockIdx.x & 15) * 16;       // N tile (16 cols of 256)
    const int mBlk = blockIdx.x >> 4;              // 0..31 (512 rows each)
    const int mrow = mBlk * 512 + wave * 64;
    const int m = lane & 15, h = lane >> 4, n = lane & 15;
    v8f c0 = {}, c1 = {}, c2 = {}, c3 = {};

    for (int ph = 0; ph < 2; ++ph) {
        __syncthreads();
        for (int i = tid; i < 512 * 16; i += 256) {
            int k = i >> 4, nn = i & 15;
            Bs2[(k >> 1) * 32 + nn * 2 + (k & 1)] =
                W[(size_t)(ph * 512 + k) * N_OUT + nb + nn];
        }
        __syncthreads();
        for (int k0 = 0; k0 < 512; k0 += 4) {
            const int gk = ph * 512 + k0;
            if (k0 + 64 < 512)
                __builtin_prefetch(&yf[(size_t)(mrow + m) * D_INNER + gk + 64], 0, 3);
            v2f b = load_b_lds(Bs2, k0, lane);
            v2f a0, a1, a2, a3;
#pragma unroll
            for (int ti = 0; ti < 4; ++ti) {
                const size_t p = (size_t)(mrow + ti * 16 + m) * D_INNER + gk + 2 * h;
                v2f a;
                a.x = (yf[p] + yb[p]) * g[p];
                a.y = (yf[p + 1] + yb[p + 1]) * g[p + 1];
                if (ti == 0) a0 = a; else if (ti == 1) a1 = a;
                else if (ti == 2) a2 = a; else a3 = a;
            }
            c0 = wmma4(a0, b, c0);
            c1 = wmma4(a1, b, c1);
            c2 = wmma4(a2, b, c2);
            c3 = wmma4(a3, b, c3);
        }
    }
    v8f cc[4] = {c0, c1, c2, c3};
#pragma unroll
    for (int ti = 0; ti < 4; ++ti)
#pragma unroll
        for (int r = 0; r < 8; ++r)
            out[(size_t)(mrow + ti * 16 + r + 8 * h) * N_OUT + nb + n] = cc[ti][r];
}

// ---------------------------------------------------------------------------
extern "C" void kernel_launch(void* const* d_in, const int* in_sizes, int n_in,
                              void* d_out, int out_size, void* d_ws, size_t ws_size,
                              hipStream_t stream) {
    (void)in_sizes; (void)n_in; (void)out_size; (void)ws_size;
    const float* x      = (const float*)d_in[0];
    const float* W_in   = (const float*)d_in[1];
    const float* b_in   = (const float*)d_in[2];
    const float* W_dt1  = (const float*)d_in[3];
    const float* b_dt1  = (const float*)d_in[4];
    const float* W_dt2  = (const float*)d_in[5];
    const float* b_dt2  = (const float*)d_in[6];
    const float* A_log  = (const float*)d_in[7];
    const float* W_B    = (const float*)d_in[8];
    const float* W_C    = (const float*)d_in[9];
    const float* D_prm  = (const float*)d_in[10];
    const float* W_out  = (const float*)d_in[11];
    float* out = (float*)d_out;

    float* ws = (float*)d_ws;
    const size_t MI = (size_t)M_TOK * D_INNER;          // 16.78M floats
    const size_t MS = (size_t)M_TOK * D_STATE;          // 262144 floats
    const size_t CH = (size_t)2 * BATCH * D_INNER * NCHUNK * D_STATE; // 2.10M floats
    float* xm    = ws;
    float* gate  = xm + MI;
    float* dtb   = gate + MI;
    float* yfb   = dtb + MI;
    float* ybb   = yfb + MI;
    float* dtlow = ybb + MI;
    float* Bp    = dtlow + MS;
    float* Cp    = Bp + MS;
    float* Pbuf  = Cp + MS;
    float* Qbuf  = Pbuf + CH;
    float* Hst   = Qbuf + CH;

    dim3 blk(256);
    k_gemm_in_silu<<<4096, blk, 0, stream>>>(x, W_in, b_in, xm, gate);
    k_proj16<<<384, blk, 0, stream>>>(xm, W_dt1, b_dt1, W_B, W_C, dtlow, Bp, Cp);
    k_dt_proj<<<8192, blk, 0, stream>>>(dtlow, W_dt2, b_dt2, dtb);
    k_scan_chunks<<<512, blk, 0, stream>>>(xm, dtb, Bp, A_log, Pbuf, Qbuf);
    k_scan_compose<<<32, blk, 0, stream>>>(Pbuf, Qbuf, Hst);
    k_scan_apply<<<512, blk, 0, stream>>>(xm, dtb, Bp, Cp, A_log, D_prm, Hst, yfb, ybb);
    k_gemm_out<<<512, blk, 0, stream>>>(yfb, ybb, gate, W_out, out);
}